// SimpleBlock_88278757802285
// MI455X (gfx1250) — compile-verified
//
#include <hip/hip_runtime.h>

typedef float v2f __attribute__((ext_vector_type(2)));
typedef float v8f __attribute__((ext_vector_type(8)));

#define NPTS 200000
#define KNBR 16
#define CIN 64
#define DOUT 64
#define KDIM1 68   /* 67 input features padded to 68 (multiple of 4) */
#define WAVES 4
#define TPB (WAVES * 32)

__device__ __forceinline__ float wave_max32(float v) {
#pragma unroll
  for (int off = 16; off >= 1; off >>= 1)
    v = fmaxf(v, __shfl_xor(v, off, 32));
  return v;
}

// PHASE 0: matmul1 -> accumulate BN1 sums
// PHASE 1: matmul1 -> bn1+relu -> matmul2 -> accumulate BN2 sums
// PHASE 2: matmul1 -> bn1+relu -> matmul2 -> bn2+relu -> max over K -> out
template <int PHASE>
__global__ __launch_bounds__(TPB) void pointnet_kernel(
    const float* __restrict__ p, const float* __restrict__ x,
    const int* __restrict__ idx, const float* __restrict__ w1,
    const float* __restrict__ b1, const float* __restrict__ w2,
    const float* __restrict__ b2, float* __restrict__ ws,
    float* __restrict__ out) {
  // Transposed weight tiles: w1t[d][k], k reordered so cols 0..63 = x-features
  // (w1 rows 3..66), cols 64..66 = pj (w1 rows 0..2), col 67 = zero pad.
  __shared__ float w1t[DOUT][KDIM1];
  __shared__ float w2t[DOUT][KDIM1];
  __shared__ float hbuf[WAVES][KNBR][KDIM1];
  __shared__ float b1s[DOUT], b2s[DOUT];
  __shared__ float sc1s[DOUT], sh1s[DOUT], sc2s[DOUT], sh2s[DOUT];
  __shared__ float sAcc[DOUT], qAcc[DOUT];

  const int tid = threadIdx.x;
  for (int e = tid; e < DOUT * KDIM1; e += TPB) {
    int d = e / KDIM1, k = e % KDIM1;
    float v;
    if (k < 64)      v = w1[(3 + k) * DOUT + d];
    else if (k < 67) v = w1[(k - 64) * DOUT + d];
    else             v = 0.f;
    w1t[d][k] = v;
    if (PHASE >= 1) w2t[d][k] = (k < 64) ? w2[k * DOUT + d] : 0.f;
  }
  if (tid < DOUT) {
    b1s[tid] = b1[tid];
    if (PHASE >= 1) {
      b2s[tid] = b2[tid];
      sc1s[tid] = ws[128 + tid];
      sh1s[tid] = ws[192 + tid];
    }
    if (PHASE == 2) {
      sc2s[tid] = ws[384 + tid];
      sh2s[tid] = ws[448 + tid];
    }
    if (PHASE <= 1) { sAcc[tid] = 0.f; qAcc[tid] = 0.f; }
  }
  __syncthreads();

  const int wave = tid >> 5;
  const int lane = tid & 31;
  const int n = blockIdx.x * WAVES + wave;

  if (n < NPTS) {
    // ---------------- gather + pj normalization ----------------
    const float px = p[n * 3 + 0], py = p[n * 3 + 1], pz = p[n * 3 + 2];
    float dx = 0.f, dy = 0.f, dz = 0.f, l2 = 0.f;
    if (lane < KNBR) {
      int j = idx[n * KNBR + lane];
      dx = p[j * 3 + 0] - px;
      dy = p[j * 3 + 1] - py;
      dz = p[j * 3 + 2] - pz;
      l2 = sqrtf(dx * dx + dy * dy + dz * dz);
    }
    const float rmax = wave_max32(l2);
    const float inv = 1.f / (rmax + 1e-8f);
    if (lane < KNBR) {
      hbuf[wave][lane][64] = dx * inv;
      hbuf[wave][lane][65] = dy * inv;
      hbuf[wave][lane][66] = dz * inv;
      hbuf[wave][lane][67] = 0.f;
    }
    {
      // each lane: half a neighbor row (32 floats) as 8x float4 (L2 hits)
      int j = lane >> 1;
      int cb = (lane & 1) * 32;
      int nb = idx[n * KNBR + j];
      const float4* src = (const float4*)(x + (size_t)nb * CIN + cb);
      float4* dst = (float4*)&hbuf[wave][j][cb];
#pragma unroll
      for (int i = 0; i < 8; ++i) dst[i] = src[i];
    }

    const int row = lane & 15;           // A row / B column-in-tile / D column
    const int kad = (lane >> 4) << 1;    // lanes 16..31 hold K+2,K+3
    const v8f zero8 = {0.f, 0.f, 0.f, 0.f, 0.f, 0.f, 0.f, 0.f};

    // ---------------- matmul1: (16x68) x (68x64) ----------------
    v8f acc[4];
#pragma unroll
    for (int t = 0; t < 4; ++t) acc[t] = zero8;
    for (int k0 = 0; k0 < KDIM1; k0 += 4) {
      int kk = k0 + kad;
      v2f av = *(const v2f*)&hbuf[wave][row][kk];
#pragma unroll
      for (int t = 0; t < 4; ++t) {
        v2f bv = *(const v2f*)&w1t[t * 16 + row][kk];
        acc[t] = __builtin_amdgcn_wmma_f32_16x16x4_f32(
            false, av, false, bv, (short)0, acc[t], false, false);
      }
    }

    if (PHASE == 0) {
#pragma unroll
      for (int t = 0; t < 4; ++t) {
        int c = t * 16 + row;
        float bias = b1s[c];
        float s = 0.f, q = 0.f;
#pragma unroll
        for (int v = 0; v < 8; ++v) {
          float a = acc[t][v] + bias;
          s += a;
          q += a * a;
        }
        atomicAdd(&sAcc[c], s);
        atomicAdd(&qAcc[c], q);
      }
    } else {
      // bn1 + relu, write y back into hbuf cols 0..63 (A layout for matmul2)
#pragma unroll
      for (int t = 0; t < 4; ++t) {
        int c = t * 16 + row;
        float bias = b1s[c], sc = sc1s[c], sh = sh1s[c];
#pragma unroll
        for (int v = 0; v < 8; ++v) {
          float a = fmaxf((acc[t][v] + bias) * sc + sh, 0.f);
          hbuf[wave][v + 8 * (lane >> 4)][c] = a;
        }
      }
      // ---------------- matmul2: (16x64) x (64x64) ----------------
      v8f acc2[4];
#pragma unroll
      for (int t = 0; t < 4; ++t) acc2[t] = zero8;
      for (int k0 = 0; k0 < 64; k0 += 4) {
        int kk = k0 + kad;
        v2f av = *(const v2f*)&hbuf[wave][row][kk];
#pragma unroll
        for (int t = 0; t < 4; ++t) {
          v2f bv = *(const v2f*)&w2t[t * 16 + row][kk];
          acc2[t] = __builtin_amdgcn_wmma_f32_16x16x4_f32(
              false, av, false, bv, (short)0, acc2[t], false, false);
        }
      }
      if (PHASE == 1) {
#pragma unroll
        for (int t = 0; t < 4; ++t) {
          int c = t * 16 + row;
          float bias = b2s[c];
          float s = 0.f, q = 0.f;
#pragma unroll
          for (int v = 0; v < 8; ++v) {
            float a = acc2[t][v] + bias;
            s += a;
            q += a * a;
          }
          atomicAdd(&sAcc[c], s);
          atomicAdd(&qAcc[c], q);
        }
      } else {
        // bn2 + relu + max over K rows, write output
#pragma unroll
        for (int t = 0; t < 4; ++t) {
          int c = t * 16 + row;
          float bias = b2s[c], sc = sc2s[c], sh = sh2s[c];
          float mx = 0.f;  // relu output >= 0
#pragma unroll
          for (int v = 0; v < 8; ++v) {
            float z = fmaxf((acc2[t][v] + bias) * sc + sh, 0.f);
            mx = fmaxf(mx, z);
          }
          mx = fmaxf(mx, __shfl_xor(mx, 16, 32));  // rows 0-7 vs 8-15
          if (lane < 16) out[(size_t)n * DOUT + c] = mx;
        }
      }
    }
  }

  if (PHASE <= 1) {
    __syncthreads();
    if (tid < DOUT) {
      const int base = (PHASE == 0) ? 0 : 256;
      atomicAdd(&ws[base + tid], sAcc[tid]);
      atomicAdd(&ws[base + 64 + tid], qAcc[tid]);
    }
  }
}

__global__ void finalize_kernel(const float* __restrict__ sums,
                                const float* __restrict__ g,
                                const float* __restrict__ be,
                                float* __restrict__ scale,
                                float* __restrict__ shift) {
  int c = threadIdx.x;
  if (c < DOUT) {
    const float cnt = (float)NPTS * (float)KNBR;
    float m = sums[c] / cnt;
    float var = sums[64 + c] / cnt - m * m;
    float sc = g[c] * rsqrtf(var + 1e-5f);
    scale[c] = sc;
    shift[c] = be[c] - m * sc;
  }
}

extern "C" void kernel_launch(void* const* d_in, const int* in_sizes, int n_in,
                              void* d_out, int out_size, void* d_ws,
                              size_t ws_size, hipStream_t stream) {
  const float* p   = (const float*)d_in[0];
  const float* x   = (const float*)d_in[1];
  // d_in[2] = o (unused)
  const int*   idx = (const int*)d_in[3];
  const float* w1  = (const float*)d_in[4];
  const float* b1  = (const float*)d_in[5];
  const float* g1  = (const float*)d_in[6];
  const float* be1 = (const float*)d_in[7];
  const float* w2  = (const float*)d_in[8];
  const float* b2  = (const float*)d_in[9];
  const float* g2  = (const float*)d_in[10];
  const float* be2 = (const float*)d_in[11];
  float* out = (float*)d_out;
  float* ws  = (float*)d_ws;

  // ws float layout: [0]sum1 [64]sq1 [128]scale1 [192]shift1
  //                  [256]sum2 [320]sq2 [384]scale2 [448]shift2
  hipMemsetAsync(ws, 0, 512 * sizeof(float), stream);

  dim3 grid((NPTS + WAVES - 1) / WAVES), block(TPB);
  pointnet_kernel<0><<<grid, block, 0, stream>>>(p, x, idx, w1, b1, w2, b2, ws, out);
  finalize_kernel<<<1, 64, 0, stream>>>(ws + 0, g1, be1, ws + 128, ws + 192);
  pointnet_kernel<1><<<grid, block, 0, stream>>>(p, x, idx, w1, b1, w2, b2, ws, out);
  finalize_kernel<<<1, 64, 0, stream>>>(ws + 256, g2, be2, ws + 384, ws + 448);
  pointnet_kernel<2><<<grid, block, 0, stream>>>(p, x, idx, w1, b1, w2, b2, ws, out);
}